// Blur_21930103014038
// MI455X (gfx1250) — compile-verified
//
#include <hip/hip_runtime.h>
#include <cstdint>

// Blur (upfirdn2d, up=down=1), x=(8,256,128,128) f32, pad=(2,1) both dims,
// 4x4 separable kernel. Memory-bound: ~268MB traffic -> ~11.5us at 23.3TB/s.
// Strategy: async global->LDS staging (CDNA5 ASYNCcnt path) + separable
// rolling-window compute (4 LDS reads + ~8 VALU per output).

#define BATCH   8
#define CHANS   256
#define HDIM    128
#define WDIM    128
#define TH      32            // output rows per block
#define NR      (TH + 3)      // staged input rows (halo: 2 above, 1 below)
#define LW      136           // LDS row width in floats (4 pad | 128 data | 4 pad)
#define PLANE_F (HDIM * WDIM) // 16384 floats per (b,c) plane

__global__ void __launch_bounds__(256)
blur_tile_kernel(const float* __restrict__ x,
                 const float* __restrict__ K,   // 16 floats, [4][4]
                 float* __restrict__ out)
{
    __shared__ float tile[NR * LW];

    const int tid   = threadIdx.x;      // 0..255
    const int lane  = tid & 31;
    const int wave  = tid >> 5;         // 0..7
    const int chunk = blockIdx.x;       // 0..3   (row chunk)
    const int plane = blockIdx.y;       // 0..2047 (b*C + c)
    const int r0    = chunk * TH;

    const float* src = x   + ((size_t)plane << 14);
    float*       dst = out + ((size_t)plane << 14);

    // ---------------- Stage 1: fill LDS tile ----------------
    // Each wave handles rows {wave, wave+8, ...}. One wave32 covers a full
    // 128-float row with one 16B async load per lane.
    for (int row = wave; row < NR; row += 8) {
        const int g = r0 - 2 + row;                 // global input row
        float* lrow = &tile[row * LW];
        if (lane == 0) { lrow[0] = 0.f; lrow[1] = 0.f; lrow[2] = 0.f; lrow[3] = 0.f; }
        if (lane == 1) { lrow[132] = 0.f; lrow[133] = 0.f; lrow[134] = 0.f; lrow[135] = 0.f; }
        if ((unsigned)g < (unsigned)HDIM) {
            const float* gp = src + g * WDIM + (lane << 2);
            const unsigned long long ga = (unsigned long long)(uintptr_t)gp;
            // Flat address of an LDS location: low 32 bits == LDS byte offset.
            const uint32_t la = (uint32_t)(uintptr_t)(lrow + 4 + (lane << 2));
            asm volatile("global_load_async_to_lds_b128 %0, %1, off"
                         :: "v"(la), "v"(ga) : "memory");
        } else {
            // out-of-plane row -> zeros (plain ds_store, 16B aligned)
            float4 z = make_float4(0.f, 0.f, 0.f, 0.f);
            *(float4*)(lrow + 4 + (lane << 2)) = z;
        }
    }
    asm volatile("s_wait_asynccnt 0" ::: "memory");
    __syncthreads();

    // ---------------- Weights (generic, flipped, separable) ----------------
    // Reference kernel is rank-1 and normalized: K[i][j] == rowsum_i*colsum_j/total.
    float cs0, cs1, cs2, cs3, rs0, rs1, rs2, rs3;
    {
        float k[16];
        #pragma unroll
        for (int i = 0; i < 16; ++i) k[i] = K[i];   // uniform -> scalar loads
        cs0 = k[0] + k[1] + k[2] + k[3];
        cs1 = k[4] + k[5] + k[6] + k[7];
        cs2 = k[8] + k[9] + k[10] + k[11];
        cs3 = k[12] + k[13] + k[14] + k[15];
        rs0 = k[0] + k[4] + k[8]  + k[12];
        rs1 = k[1] + k[5] + k[9]  + k[13];
        rs2 = k[2] + k[6] + k[10] + k[14];
        rs3 = k[3] + k[7] + k[11] + k[15];
        const float inv = 1.0f / (cs0 + cs1 + cs2 + cs3);
        rs0 *= inv; rs1 *= inv; rs2 *= inv; rs3 *= inv;
    }
    // True convolution => flipped kernel: w[i][j] = K[3-i][3-j] = cs[3-i]*rs[3-j]
    const float cw0 = cs3, cw1 = cs2, cw2 = cs1, cw3 = cs0;
    const float rw0 = rs3, rw1 = rs2, rw2 = rs1, rw3 = rs0;

    // ---------------- Stage 2: separable compute ----------------
    // Thread -> (column, half): each thread computes 16 rows of one column,
    // rolling a 4-deep window of horizontal FIR sums in registers.
    const int xcol = tid & 127;
    const int half = tid >> 7;            // 0 or 1
    const int lr0  = half * (TH / 2);     // local start row in tile space

    #define HSUM(q) ({ const float* _p = &tile[(q) * LW + xcol + 2];            \
                       rw0*_p[0] + rw1*_p[1] + rw2*_p[2] + rw3*_p[3]; })

    float h0 = HSUM(lr0 + 0);
    float h1 = HSUM(lr0 + 1);
    float h2 = HSUM(lr0 + 2);

    float* orow = dst + (r0 + lr0) * WDIM + xcol;
    #pragma unroll
    for (int i = 0; i < TH / 2; ++i) {
        const float h3 = HSUM(lr0 + 3 + i);
        orow[i * WDIM] = cw0 * h0 + cw1 * h1 + cw2 * h2 + cw3 * h3;
        h0 = h1; h1 = h2; h2 = h3;
    }
    #undef HSUM
}

extern "C" void kernel_launch(void* const* d_in, const int* in_sizes, int n_in,
                              void* d_out, int out_size, void* d_ws, size_t ws_size,
                              hipStream_t stream) {
    (void)in_sizes; (void)n_in; (void)out_size; (void)d_ws; (void)ws_size;
    const float* x = (const float*)d_in[0];
    const float* K = (const float*)d_in[1];
    float* out = (float*)d_out;

    dim3 grid(HDIM / TH, BATCH * CHANS);   // (4, 2048)
    blur_tile_kernel<<<grid, 256, 0, stream>>>(x, K, out);
}